// CrossGraphConvolution_34961033789910
// MI455X (gfx1250) — compile-verified
//
#include <hip/hip_runtime.h>
#include <math.h>

typedef __attribute__((ext_vector_type(2))) float v2f;
typedef __attribute__((ext_vector_type(4))) float v4f;
typedef __attribute__((ext_vector_type(8))) float v8f;

#define NPG 32
#define DIN 128
#define NOUT 128
#define EPS 1e-6f
#define XS 132   // padded LDS row stride for [32][128] arrays (bank-conflict free)
#define CS 33    // padded LDS row stride for C [32][32]

// Fused CrossGraphConvolution: one workgroup per graph pair.
__global__ void __launch_bounds__(256)
cross_graph_conv_kernel(const float* __restrict__ x_left,
                        const float* __restrict__ x_right,
                        const float* __restrict__ weight,
                        float* __restrict__ out1,
                        float* __restrict__ out2)
{
    extern __shared__ __align__(16) float smem[];
    float* Xl = smem;                  // 32 * XS
    float* Xr = Xl + NPG * XS;
    float* Gl = Xr + NPG * XS;
    float* Gr = Gl + NPG * XS;
    float* Cm = Gr + NPG * XS;         // 32 * CS
    float* nl = Cm + NPG * CS;         // 32
    float* nr = nl + NPG;              // 32
    float* invrow = nr + NPG;          // 32
    float* invcol = invrow + NPG;      // 32

    const int b    = blockIdx.x;
    const int tid  = threadIdx.x;
    const int lane = tid & 31;
    const int wave = tid >> 5;
    const int half = lane >> 4;        // 0: K-off {0,1}, 1: K-off {2,3}
    const int l15  = lane & 15;

    // ---- Stage 0: stage Xl, Xr tiles into LDS (16B vector loads) ----
    for (int t = tid; t < 2 * NPG * (DIN / 4); t += 256) {
        int mat = t >> 10;             // NPG*DIN/4 == 1024 per matrix
        int rem = t & 1023;
        int r   = rem >> 5;            // 32 float4 per row
        int c4  = rem & 31;
        const float* src = mat ? x_right : x_left;
        float* dst       = mat ? Xr : Xl;
        v4f v = *(const v4f*)(src + (size_t)b * NPG * DIN + r * DIN + c4 * 4);
        *(v4f*)(dst + r * XS + c4 * 4) = v;
    }
    __syncthreads();

    // ---- Stage 1: row norms ----
    if (tid < 2 * NPG) {
        const float* row = (tid < NPG) ? (Xl + tid * XS) : (Xr + (tid - NPG) * XS);
        float s = 0.f;
        for (int d = 0; d < DIN; ++d) { float v = row[d]; s += v * v; }
        float n = sqrtf(s);
        if (tid < NPG) nl[tid] = n; else nr[tid - NPG] = n;
    }
    __syncthreads();

    // ---- Stage 2: S = Xl * Xr^T  (shared by both directions), WMMA f32 ----
    if (wave < 4) {                    // waves 0..3: one 16x16 tile each
        int m0 = (wave >> 1) * 16;
        int n0 = (wave & 1) * 16;
        const float* arow = Xl + (m0 + l15) * XS + half * 2;
        const float* brow = Xr + (n0 + l15) * XS + half * 2;   // B[k][n] = Xr[n][k]
        v8f acc = {};
        for (int k0 = 0; k0 < DIN; k0 += 4) {
            v2f a  = { arow[k0], arow[k0 + 1] };
            v2f bb = { brow[k0], brow[k0 + 1] };
            acc = __builtin_amdgcn_wmma_f32_16x16x4_f32(false, a, false, bb,
                                                        (short)0, acc, false, false);
        }
        for (int v = 0; v < 8; ++v)
            Cm[(m0 + v + half * 8) * CS + (n0 + l15)] = acc[v];
    }
    __syncthreads();

    // ---- Stage 2b: C = relu(dot / max(|xl||xr|, eps)) ----
    for (int t = tid; t < NPG * NPG; t += 256) {
        int i = t >> 5, j = t & 31;
        float nrm = fmaxf(nl[i] * nr[j], EPS);
        Cm[i * CS + j] = fmaxf(Cm[i * CS + j] / nrm, 0.f);
    }
    __syncthreads();

    // ---- Stage 3: row/col sums (+ NPG*eps), reciprocal ----
    if (tid < NPG) {
        float s = NPG * EPS;
        for (int j = 0; j < NPG; ++j) s += Cm[tid * CS + j];
        invrow[tid] = 1.f / s;
    } else if (tid < 2 * NPG) {
        int j = tid - NPG;
        float s = NPG * EPS;
        for (int i = 0; i < NPG; ++i) s += Cm[i * CS + j];
        invcol[j] = 1.f / s;
    }
    __syncthreads();

    // ---- Stage 4: Gl = rownorm(C)*Xr ; Gr = colnorm(C)^T*Xl  (WMMA, K=32) ----
    for (int q = 0; q < 4; ++q) {
        int tt   = wave * 4 + q;       // 0..31 tiles
        int side = tt >> 4;            // 0: left targets, 1: right targets
        int r    = tt & 15;
        int m0   = (r >> 3) * 16;
        int n0   = (r & 7) * 16;
        const float* Xsrc = side ? Xl : Xr;
        float* g          = side ? Gr : Gl;
        int m    = m0 + l15;
        float inv = side ? invcol[m] : invrow[m];
        v8f acc = {};
        for (int k0 = 0; k0 < NPG; k0 += 4) {
            int k = k0 + half * 2;
            v2f a, bb;
            if (side == 0)
                a = (v2f){ Cm[m * CS + k] * inv, Cm[m * CS + k + 1] * inv };
            else
                a = (v2f){ Cm[k * CS + m] * inv, Cm[(k + 1) * CS + m] * inv };
            bb = (v2f){ Xsrc[k * XS + n0 + l15], Xsrc[(k + 1) * XS + n0 + l15] };
            acc = __builtin_amdgcn_wmma_f32_16x16x4_f32(false, a, false, bb,
                                                        (short)0, acc, false, false);
        }
        for (int v = 0; v < 8; ++v)
            g[(m0 + v + half * 8) * XS + n0 + l15] = acc[v];
    }
    __syncthreads();

    // ---- Stage 5: fused triple GEMM vs w^2 + cosine finalize ----
    // Virtual A rows: p = x*g, q = x*x, r = g*g  (64 nodes: 0..31 left, 32..63 right)
    for (int q = 0; q < 4; ++q) {
        int tt = wave * 4 + q;         // 0..31 output tiles of [64 x 128]
        int m0 = (tt >> 3) * 16;       // node-row block
        int n0 = (tt & 7) * 16;        // out-channel block
        int node = m0 + l15;           // all lanes same side within a tile
        const float* xp;
        const float* gp;
        if (node < NPG) { xp = Xl + node * XS;          gp = Gl + node * XS; }
        else            { xp = Xr + (node - NPG) * XS;  gp = Gr + (node - NPG) * XS; }
        const float* wrow = weight + (size_t)(n0 + l15) * DIN;   // B[k][n] = w[n][k]^2
        v8f accN = {}, accT = {}, accG = {};
        for (int k0 = 0; k0 < DIN; k0 += 4) {
            int k = k0 + half * 2;
            float x0 = xp[k], x1 = xp[k + 1];
            float g0 = gp[k], g1 = gp[k + 1];
            float w0 = wrow[k], w1 = wrow[k + 1];
            v2f bb = { w0 * w0, w1 * w1 };
            v2f ap = { x0 * g0, x1 * g1 };
            v2f aq = { x0 * x0, x1 * x1 };
            v2f ar = { g0 * g0, g1 * g1 };
            accN = __builtin_amdgcn_wmma_f32_16x16x4_f32(false, ap, false, bb, (short)0, accN, false, false);
            accT = __builtin_amdgcn_wmma_f32_16x16x4_f32(false, aq, false, bb, (short)0, accT, false, false);
            accG = __builtin_amdgcn_wmma_f32_16x16x4_f32(false, ar, false, bb, (short)0, accG, false, false);
        }
        for (int v = 0; v < 8; ++v) {
            int m = m0 + v + half * 8;
            int o = n0 + l15;
            float dt  = sqrtf(accT[v] + EPS);
            float dg  = sqrtf(accG[v] + EPS);
            float val = accN[v] / fmaxf(dt * dg, EPS);
            if (m < NPG) out1[((size_t)b * NPG + m) * NOUT + o] = val;
            else         out2[((size_t)b * NPG + (m - NPG)) * NOUT + o] = val;
        }
    }
}

extern "C" void kernel_launch(void* const* d_in, const int* in_sizes, int n_in,
                              void* d_out, int out_size, void* d_ws, size_t ws_size,
                              hipStream_t stream) {
    (void)n_in; (void)d_ws; (void)ws_size; (void)out_size;
    // setup_inputs order: x_left, batch_left, x_right, batch_right,
    //                     edge_row, edge_col, weight
    const float* x_left  = (const float*)d_in[0];
    const float* x_right = (const float*)d_in[2];
    const float* weight  = (const float*)d_in[6];

    const int N = in_sizes[0] / DIN;         // 4096 nodes per side
    const int B = N / NPG;                   // 128 graph pairs

    float* out1 = (float*)d_out;
    float* out2 = out1 + (size_t)N * NOUT;

    const size_t lds_bytes =
        (size_t)(4 * NPG * XS + NPG * CS + 4 * NPG) * sizeof(float);  // ~72 KB

    cross_graph_conv_kernel<<<dim3(B), dim3(256), lds_bytes, stream>>>(
        x_left, x_right, weight, out1, out2);
}